// Retriever_39367670235663
// MI455X (gfx1250) — compile-verified
//
#include <hip/hip_runtime.h>
#include <hip/hip_bf16.h>
#include <math.h>

typedef __attribute__((ext_vector_type(2))) float v2f;
typedef __attribute__((ext_vector_type(8))) float v8f;
typedef __attribute__((ext_vector_type(4))) int   v4i;

#define GROUPS 12
#define POOL   64
#define TOPK   8
#define LAYERS 24
#define RANK   64
#define HID    1024
#define NBATCH 64
#define QDIM   768                       // GROUPS * 64
#define CH     64                        // channels per group
#define D1     (HID * RANK * LAYERS)     // per-slot (a or b) flat size = 1572864

// ---------------------------------------------------------------------------
// CDNA5 async global->LDS copy (ASYNCcnt-tracked, no VGPR round trip).
// ---------------------------------------------------------------------------
#if __has_builtin(__builtin_amdgcn_global_load_async_to_lds_b128)
#define HAVE_ASYNC_COPY 1
#else
#define HAVE_ASYNC_COPY 0
#endif

#if HAVE_ASYNC_COPY
typedef __attribute__((address_space(1))) v4i* gv4i_p;
typedef __attribute__((address_space(3))) v4i* lv4i_p;

__device__ __forceinline__ void async_copy_b128(const float* g, float* l) {
    __builtin_amdgcn_global_load_async_to_lds_b128(
        (gv4i_p)(void*)g, (lv4i_p)(void*)l, 0, 0);
}
#if __has_builtin(__builtin_amdgcn_s_wait_asynccnt)
#define WAIT_ASYNC(N) __builtin_amdgcn_s_wait_asynccnt(N)
#else
#define WAIT_ASYNC(N) asm volatile("s_wait_asynccnt %0" :: "n"(N) : "memory")
#endif
#endif

// ---------------------------------------------------------------------------
// Kernel A: routing. One block of 768 threads.
// ---------------------------------------------------------------------------
__global__ __launch_bounds__(768)
void route_kernel(const float* __restrict__ q,      // [64, 768]
                  const float* __restrict__ keys,   // [1, 12, 64, 64]
                  int* __restrict__ idx_out,        // [8]
                  float* __restrict__ w_out)        // [8]
{
    __shared__ float rnorm[NBATCH * GROUPS];   // 1/||q_{b,g}||
    __shared__ float S[GROUPS * CH];           // sum_b qn[b,g,c]
    __shared__ float msim[POOL];

    const int t = threadIdx.x;   // 0..767

    // phase 0: per-(b,g) inverse norms
    {
        const int b = t / GROUPS;
        const int g = t % GROUPS;
        const float* row = q + b * QDIM + g * CH;
        float s = 0.f;
        for (int c = 0; c < CH; ++c) { float v = row[c]; s += v * v; }
        rnorm[t] = 1.0f / fmaxf(sqrtf(s), 1e-8f);
    }
    __syncthreads();

    // phase 1: S[g][c] = sum_b q[b,g,c] * rnorm[b,g]
    {
        const int g = t >> 6;      // 0..11
        const int c = t & 63;
        float s = 0.f;
        for (int b = 0; b < NBATCH; ++b)
            s += q[b * QDIM + g * CH + c] * rnorm[b * GROUPS + g];
        S[t] = s;
    }
    if (t < POOL) msim[t] = 0.f;
    __syncthreads();

    // phase 2: msim[p] += dot(S[g], kn[g,p]) for each g (scaled later)
    {
        const int g = t >> 6;      // 0..11
        const int p = t & 63;
        const float* krow = keys + (g * POOL + p) * CH;
        float dot = 0.f, knrm = 0.f;
        for (int c = 0; c < CH; ++c) {
            float kv = krow[c];
            dot  += S[g * CH + c] * kv;
            knrm += kv * kv;
        }
        atomicAdd(&msim[p], dot / fmaxf(sqrtf(knrm), 1e-8f));
    }
    __syncthreads();

    // phase 3: serial top-8 (64 entries, trivial)
    if (t == 0) {
        const float scale = 1.0f / (float)(NBATCH * GROUPS);
        float vals[TOPK];
        int   idxs[TOPK];
        float wsum = 0.f;
        for (int k = 0; k < TOPK; ++k) {
            float best = -INFINITY; int bi = 0;
            for (int p = 0; p < POOL; ++p) {
                float m = msim[p] * scale;
                if (m > best) { best = m; bi = p; }
            }
            msim[bi] = -INFINITY;
            vals[k] = best; idxs[k] = bi;
            wsum += best;
        }
        const float inv = 1.0f / (wsum + 1e-9f);
        for (int k = 0; k < TOPK; ++k) {
            idx_out[k] = idxs[k];
            w_out[k]   = vals[k] * inv;
        }
    }
}

// ---------------------------------------------------------------------------
// Kernel B: out[l,x,y] = sum_n w[n] * sum_r A[n,l,r,x] * B[n,l,r,y]
// Workgroup tile 128(M) x 256(N), K = RANK = 64, 8 waves of 64x64 each,
// f32 WMMA 16x16x4. Double-buffered async global->LDS pipeline:
//   stage(n+1) issued async, s_wait_asynccnt <= 24 retires stage(n),
//   weight w[n] applied to the A fragments via v_pk_mul_f32,
//   streaming (non-temporal) output stores.
// ---------------------------------------------------------------------------
#define BM   128
#define BN   256
#define BK   64
#define LDA  (BM + 4)   // 132 floats: half-wave K-split lands on distinct banks
#define LDB  (BN + 4)   // 260 floats
#define STAGE_OPS 24    // async-b128 instructions per wave per stage (8 A + 16 B)

__device__ __forceinline__ void stage_tiles(const float* __restrict__ Ag,
                                            const float* __restrict__ Bg,
                                            float* __restrict__ As,
                                            float* __restrict__ Bs,
                                            int t)
{
#if HAVE_ASYNC_COPY
    #pragma unroll
    for (int i = 0; i < 8; ++i) {            // A: 64 x 128 f32 = 2048 b128
        int id = t + i * 256;
        int r  = id >> 5;                    // 32 float4 per row
        int c4 = id & 31;
        async_copy_b128(Ag + (size_t)r * HID + c4 * 4, &As[r * LDA + c4 * 4]);
    }
    #pragma unroll
    for (int i = 0; i < 16; ++i) {           // B: 64 x 256 f32 = 4096 b128
        int id = t + i * 256;
        int r  = id >> 6;                    // 64 float4 per row
        int c4 = id & 63;
        async_copy_b128(Bg + (size_t)r * HID + c4 * 4, &Bs[r * LDB + c4 * 4]);
    }
#else
    #pragma unroll
    for (int i = 0; i < 8; ++i) {
        int id = t + i * 256;
        int r  = id >> 5;
        int c4 = id & 31;
        *reinterpret_cast<float4*>(&As[r * LDA + c4 * 4]) =
            *reinterpret_cast<const float4*>(Ag + (size_t)r * HID + c4 * 4);
    }
    #pragma unroll
    for (int i = 0; i < 16; ++i) {
        int id = t + i * 256;
        int r  = id >> 6;
        int c4 = id & 63;
        *reinterpret_cast<float4*>(&Bs[r * LDB + c4 * 4]) =
            *reinterpret_cast<const float4*>(Bg + (size_t)r * HID + c4 * 4);
    }
#endif
}

__global__ __launch_bounds__(256)
void lowrank_gemm_kernel(const float* __restrict__ wo,    // [64, 2, D1]
                         const int*   __restrict__ sel,   // [8]
                         const float* __restrict__ selw,  // [8]
                         float* __restrict__ out)         // [24, 1024, 1024]
{
    __shared__ float As[2][BK * LDA];   // 2 x 33,792 B
    __shared__ float Bs[2][BK * LDB];   // 2 x 66,560 B  (~196 KB of 320 KB/WGP)

    const int l  = blockIdx.z;
    const int m0 = blockIdx.x * BM;
    const int n0 = blockIdx.y * BN;

    const int t     = threadIdx.x;
    const int lane  = t & 31;
    const int wave  = t >> 5;            // 0..7
    const int wm    = (wave & 1) * 64;   // wave M offset inside tile
    const int wn    = (wave >> 1) * 64;  // wave N offset inside tile
    const int l15   = lane & 15;
    const int khalf = (lane >> 4) * 2;   // lanes 16-31 handle K+2,K+3

    v8f acc[4][4] = {};

    const size_t lbase = (size_t)(l * RANK) * HID;

    // prologue: stage n=0 into buffer 0
    {
        const size_t pbase = (size_t)sel[0] * (2u * (unsigned)D1);
        stage_tiles(wo + pbase + lbase + m0,
                    wo + pbase + (size_t)D1 + lbase + n0,
                    As[0], Bs[0], t);
    }

    for (int n = 0; n < TOPK; ++n) {
        const int buf = n & 1;

        // issue prefetch of stage n+1 into the other buffer
        if (n + 1 < TOPK) {
            const size_t pbase = (size_t)sel[n + 1] * (2u * (unsigned)D1);
            stage_tiles(wo + pbase + lbase + m0,
                        wo + pbase + (size_t)D1 + lbase + n0,
                        As[buf ^ 1], Bs[buf ^ 1], t);
        }

#if HAVE_ASYNC_COPY
        // retire stage n (<= STAGE_OPS outstanding leaves only stage n+1 in flight)
        if (n + 1 < TOPK) { WAIT_ASYNC(STAGE_OPS); } else { WAIT_ASYNC(0); }
#endif
        __syncthreads();   // stage n visible to all waves

        const float w = selw[n];
        const float* __restrict__ Asb = As[buf];
        const float* __restrict__ Bsb = Bs[buf];

        #pragma unroll
        for (int k = 0; k < BK; k += 4) {
            v2f af[4], bf[4];
            #pragma unroll
            for (int i = 0; i < 4; ++i) {
                const int m = wm + i * 16 + l15;
                af[i].x = Asb[(k + khalf + 0) * LDA + m];
                af[i].y = Asb[(k + khalf + 1) * LDA + m];
                af[i] *= w;                       // dis_weight fold (v_pk_mul_f32)
                const int nn = wn + i * 16 + l15;
                bf[i].x = Bsb[(k + khalf + 0) * LDB + nn];
                bf[i].y = Bsb[(k + khalf + 1) * LDB + nn];
            }
            #pragma unroll
            for (int i = 0; i < 4; ++i)
                #pragma unroll
                for (int j = 0; j < 4; ++j)
                    acc[i][j] = __builtin_amdgcn_wmma_f32_16x16x4_f32(
                        false, af[i], false, bf[j],
                        (short)0, acc[i][j], false, false);
        }

        __syncthreads();   // done reading buf before it is overwritten
    }

    // store: D layout -> VGPR v: lanes 0-15: (M=v, N=lane); lanes 16-31: (M=v+8)
    // Output is write-once streamed: use non-temporal stores (TH=NT).
    float* outl = out + (size_t)l * HID * HID;
    const int mhalf = (lane >> 4) * 8;
    #pragma unroll
    for (int i = 0; i < 4; ++i) {
        const int mrow = m0 + wm + i * 16 + mhalf;
        #pragma unroll
        for (int j = 0; j < 4; ++j) {
            const int ncol = n0 + wn + j * 16 + l15;
            #pragma unroll
            for (int v = 0; v < 8; ++v)
                __builtin_nontemporal_store(acc[i][j][v],
                    &outl[(size_t)(mrow + v) * HID + ncol]);
        }
    }
}

// ---------------------------------------------------------------------------
extern "C" void kernel_launch(void* const* d_in, const int* in_sizes, int n_in,
                              void* d_out, int out_size, void* d_ws, size_t ws_size,
                              hipStream_t stream) {
    const float* queries = (const float*)d_in[0];   // [64, 768]
    const float* keys    = (const float*)d_in[1];   // [1, 12, 64, 64]
    const float* wo      = (const float*)d_in[2];   // [64, 2, D1]
    float* out = (float*)d_out;                     // [24, 1024, 1024]

    int*   d_idx = (int*)d_ws;                      // 8 ints
    float* d_w   = (float*)((char*)d_ws + 32);      // 8 floats

    route_kernel<<<1, 768, 0, stream>>>(queries, keys, d_idx, d_w);

    dim3 grid(HID / BM, HID / BN, LAYERS);          // (8, 4, 24)
    lowrank_gemm_kernel<<<grid, 256, 0, stream>>>(wo, d_idx, d_w, out);
}